// DTW_63831803953806
// MI455X (gfx1250) — compile-verified
//
#include <hip/hip_runtime.h>
#include <stdint.h>

// DTW: x (1,1024) vs y (128,1024), fp32.
//   C[b,i,j] = (x[i]-y[b,j])^2
//   D[i,j]   = C[i,j] + min(D[i-1,j-1], D[i-1,j], D[i,j-1])   (inf borders)
//   greedy walk from (0,0): 2047 steps over {diag,down,right}, summing values.
//
// MI455X design notes:
//  - (min,+) recurrence: WMMA inapplicable; binding constraint is the 2047-step
//    serial diagonal chain + per-step sync, not FLOPs or HBM (516KB total I/O).
//  - 1 block/batch, 8 wave32 waves, 4 cols/thread; 3 rotating LDS diagonal
//    buffers; one barrier per antidiagonal.
//  - x staged via CDNA5 async global->LDS (ASYNCcnt) path.
//  - Greedy walker fused (trails wavefront by exactly 2 diagonals) -> the full
//    512MB D matrix is never materialized.
//  - Hot loop unrolled x3 with compile-time buffer indices: zero %3 SALU on the
//    critical path; t==0 specials peeled.

#define M_LEN 1024
#define N_LEN 1024
#define DIAGS (M_LEN + N_LEN - 1)   // 2047 path steps
#define DMAX  (M_LEN + N_LEN - 2)   // last antidiagonal index (2046)
#define THREADS 256
#define CPT 4                        // columns per thread
#define BUF_W (N_LEN + 8)            // 4 inf-pad front, 4 inf-pad back
                                     // column j lives at index j+4

__global__ __launch_bounds__(THREADS, 2)
void dtw_fused_kernel(const float* __restrict__ x,
                      const float* __restrict__ y,
                      float* __restrict__ out)
{
    __shared__ __align__(16) float xs[M_LEN];
    __shared__ __align__(16) float buf[3][BUF_W];

    const int   tid   = threadIdx.x;
    const int   b     = blockIdx.x;
    const int   jbase = tid * CPT;
    const float INF   = __builtin_inff();

    // ---- stage x[0:1024] into LDS via async global->LDS (CDNA5) ----
    {
        unsigned  lds_off = (unsigned)(size_t)(&xs[jbase]);
        uint64_t  gaddr   = (uint64_t)(const void*)(x + jbase);
        asm volatile("global_load_async_to_lds_b128 %0, %1, off"
                     :: "v"(lds_off), "v"(gaddr) : "memory");
    }

    // init all three diagonal buffers (incl. pads) to +inf
    for (int k = tid; k < 3 * BUF_W; k += THREADS)
        (&buf[0][0])[k] = INF;

    // my 4 y values (register-resident for the whole sweep)
    float4 y4 = *(const float4*)(y + (size_t)b * N_LEN + jbase);
    const float yv[CPT] = { y4.x, y4.y, y4.z, y4.w };

    asm volatile("s_wait_asynccnt 0" ::: "memory");
    __syncthreads();

    // r1 = my cols on diag t-1, r2 = diag t-2,
    // l1/l2 = left neighbor's value on diag t-1 / t-2, xw = rolling x window.
    float r1[CPT], r2[CPT], xw[CPT];
    #pragma unroll
    for (int c = 0; c < CPT; ++c) { r1[c] = INF; r2[c] = INF; xw[c] = 0.0f; }
    float l2 = INF;

    // greedy-path walker state (thread 0 only)
    int   wi = 0, wj = 0, wp = 0, steps = 0;
    float cost = 0.0f;

    // One antidiagonal step. CUR/PV1 are compile-time constants at every call
    // site (inlined + constant-folded), FIRST only true for the peeled t=0.
    auto dstep = [&](int t, int CUR, int PV1, bool FIRST) {
        // slide x window: cell c needs x[t - jbase - c]
        int xi = t - jbase;
        xi = xi < 0 ? 0 : (xi >= M_LEN ? M_LEN - 1 : xi);
        const float nx = xs[xi];
        xw[3] = xw[2]; xw[2] = xw[1]; xw[1] = xw[0]; xw[0] = nx;

        const float l1 = buf[PV1][jbase + 3];   // left neighbor, diag t-1

        float v[CPT];
        #pragma unroll
        for (int c = 0; c < CPT; ++c) {
            const float p1l = (c == 0) ? l1 : r1[c - 1];  // D[i,  j-1]
            const float p2l = (c == 0) ? l2 : r2[c - 1];  // D[i-1,j-1]
            float m = fminf(fminf(p1l, p2l), r1[c]);      // with D[i-1,j]
            if (FIRST && (jbase + c) == 0) m = 0.0f;      // D[0,0] = C[0,0]
            const float d   = xw[c] - yv[c];
            float       val = d * d + m;
            const int   i   = t - jbase - c;
            if (i < 0 || i >= M_LEN) val = INF;           // off-diagonal slot
            v[c] = val;
        }
        l2 = l1;
        #pragma unroll
        for (int c = 0; c < CPT; ++c) { r2[c] = r1[c]; r1[c] = v[c]; }

        *(float4*)&buf[CUR][jbase + 4] = make_float4(v[0], v[1], v[2], v[3]);
        __syncthreads();
    };

    // Fused greedy walker. Fires only when wp == t-2 (invariant: after the
    // walker at iteration t-1, wp > t-3, and each firing advances wp by >=1),
    // so its candidates live in buf[CUR] (diag t) and buf[PV1] (diag t-1).
    auto walk = [&](int t, int CUR, int PV1) {
        if (tid == 0 && wp + 2 <= t) {
            const float dv = buf[CUR][wj + 5];  // D[i+1,j+1]
            const float dn = buf[PV1][wj + 4];  // D[i+1,j  ]
            const float rt = buf[PV1][wj + 5];  // D[i,  j+1]
            float vm = dv; int k = 0;
            if (dn < vm) { vm = dn; k = 1; }
            if (rt < vm) { vm = rt; k = 2; }
            if (vm < INF) {
                cost += vm;
                wi += (k < 2);
                wj += (k != 1);
                wp  = wi + wj;
            }
            steps++;
        }
    };

    // t = 0 peeled (captures D[0,0]); walker cannot fire before t=2.
    dstep(0, 0, 2, true);
    if (tid == 0) cost = buf[0][4];

    // t = 1..2046: exactly 682 groups of 3, buffer indices cycle (1,2,0).
    for (int tb = 1; tb + 2 <= DMAX; tb += 3) {
        dstep(tb,     1, 0, false);  walk(tb,     1, 0);
        dstep(tb + 1, 2, 1, false);  walk(tb + 1, 2, 1);
        dstep(tb + 2, 0, 2, false);  walk(tb + 2, 0, 2);
    }

    // finish the walk: candidates beyond diag DMAX are +inf. After the sweep
    // the ring holds diag 2046 in buf[0], 2045 in buf[2], 2044 in buf[1],
    // consistent with (wp+k) % 3 indexing.
    if (tid == 0) {
        while (steps < DIAGS) {
            const float dv = (wp + 2 <= DMAX) ? buf[(wp + 2) % 3][wj + 5] : INF;
            const float dn = (wp + 1 <= DMAX) ? buf[(wp + 1) % 3][wj + 4] : INF;
            const float rt = (wp + 1 <= DMAX) ? buf[(wp + 1) % 3][wj + 5] : INF;
            float vm = dv; int k = 0;
            if (dn < vm) { vm = dn; k = 1; }
            if (rt < vm) { vm = rt; k = 2; }
            if (!(vm < INF)) break;   // all inf: every remaining step adds 0
            cost += vm;
            wi += (k < 2);
            wj += (k != 1);
            wp  = wi + wj;
            steps++;
        }
        out[b] = cost;
    }
}

extern "C" void kernel_launch(void* const* d_in, const int* in_sizes, int n_in,
                              void* d_out, int out_size, void* d_ws, size_t ws_size,
                              hipStream_t stream) {
    (void)n_in; (void)out_size; (void)d_ws; (void)ws_size;
    const float* x   = (const float*)d_in[0];   // (1,1024) fp32
    const float* y   = (const float*)d_in[1];   // (128,1024) fp32
    float*       out = (float*)d_out;           // (128,) fp32
    const int B = in_sizes[1] / N_LEN;          // 128
    dtw_fused_kernel<<<dim3(B), dim3(THREADS), 0, stream>>>(x, y, out);
}